// WKPooling_61168924230172
// MI455X (gfx1250) — compile-verified
//
#include <hip/hip_runtime.h>
#include <math.h>

typedef __attribute__((ext_vector_type(2))) float v2f;
typedef __attribute__((ext_vector_type(8))) float v8f;

#define NLAYERS 13
#define LAYER0  4
#define L       9     // layers used
#define BB      8
#define SS      512
#define DD      1024
#define WSZ     2
#define STRIDE  1026  // padded LDS row stride (floats)

// ---------------------------------------------------------------------------
// Kernel 1: per-token 9x9 Gram via V_WMMA_F32_16X16X4_F32, then Cholesky-based
// alpha weights + cos-variance. One block (128 thr = 4 waves) per (b,s) token.
// ---------------------------------------------------------------------------
__global__ __launch_bounds__(128) void wk_token_kernel(
    const float* __restrict__ hs,     // (13,8,512,1024) fp32
    float* __restrict__ var_out,      // (B*S)
    float* __restrict__ alpha_out)    // (B*S, 9)
{
    __shared__ float s_feat[L * STRIDE];   // 9 x 1024 tile, padded
    __shared__ float s_red[4 * 256];       // per-wave 16x16 partial Grams
    __shared__ float s_G[81];              // final 9x9 Gram

    const int tid  = threadIdx.x;
    const int lane = tid & 31;
    const int wave = tid >> 5;
    const int tok  = blockIdx.x;           // 0..4095
    const int b    = tok >> 9;
    const int s    = tok & (SS - 1);

    // Stage feat[4..12][b][s][:] into LDS (coalesced global reads).
    const size_t tok_off = ((size_t)b * SS + s) * DD;
    for (int i = tid; i < L * DD; i += 128) {
        const int l = i >> 10;
        const int c = i & (DD - 1);
        s_feat[l * STRIDE + c] =
            hs[(size_t)(LAYER0 + l) * BB * SS * DD + tok_off + c];
    }
    __syncthreads();

    // Gram = F^T F with F (1024 x 9, padded to 16 cols with copies of col 8; garbage rows
    // >= 9 only pollute D entries we never read). For this symmetric product
    // the per-lane A and B register contents coincide: lane m (m = lane&15)
    // supplies vector m elements {k0 + 2*(lane>>4), +1}.
    int vec = lane & 15;
    if (vec > L - 1) vec = L - 1;                 // clamp pad lanes to a valid row
    const int koff = (lane >> 4) << 1;
    const float* base = &s_feat[vec * STRIDE + koff];

    v8f acc = {};
    const int k_begin = wave * (DD / 4);          // 256-wide K slice per wave
    #pragma unroll 4
    for (int k0 = k_begin; k0 < k_begin + DD / 4; k0 += 4) {
        v2f ab = *(const v2f*)(base + k0);
        acc = __builtin_amdgcn_wmma_f32_16x16x4_f32(
            /*neg_a=*/false, ab, /*neg_b=*/false, ab,
            /*c_mod=*/(short)0, acc, /*reuse_a=*/false, /*reuse_b=*/false);
    }

    // Cross-wave reduction of the 16x16 accumulators -> 9x9 Gram in s_G.
    #pragma unroll
    for (int j = 0; j < 8; ++j)
        s_red[wave * 256 + lane * 8 + j] = acc[j];
    __syncthreads();

    for (int t = tid; t < 81; t += 128) {
        const int m  = t / 9, n = t % 9;
        const int rl = (m >= 8) ? (n + 16) : n;   // D layout: vgpr j = row j / j+8
        const int rj = m & 7;
        float g = 0.f;
        for (int w = 0; w < 4; ++w) g += s_red[w * 256 + rl * 8 + rj];
        s_G[t] = g;
    }
    __syncthreads();

    if (tid == 0) {
        float align9[L], nov9[L];
        for (int k = 0; k < L; ++k) {
            // window indices: full-left-only rule from the reference
            int idx[2 * WSZ + 1];
            int w = 0;
            if (k - WSZ >= 0)
                for (int i = k - WSZ; i < k; ++i) idx[w++] = i;
            int hi = k + WSZ + 1; if (hi > L) hi = L;
            for (int i = k + 1; i < hi; ++i) idx[w++] = i;
            idx[w++] = k;

            float Gw[5][5];
            for (int i = 0; i < w; ++i)
                for (int j = 0; j < w; ++j)
                    Gw[i][j] = s_G[idx[i] * 9 + idx[j]];

            // Upper Cholesky R: R^T R = Gw (matches QR's R up to row signs,
            // and the formulas below are row-sign invariant).
            float R[5][5];
            for (int i = 0; i < 5; ++i)
                for (int j = 0; j < 5; ++j) R[i][j] = 0.f;
            for (int i = 0; i < w; ++i) {
                float d2 = Gw[i][i];
                for (int p = 0; p < i; ++p) d2 -= R[p][i] * R[p][i];
                const float di = sqrtf(fmaxf(d2, 0.f));
                R[i][i] = di;
                const float inv = (di > 0.f) ? (1.f / di) : 0.f;
                for (int j = i + 1; j < w; ++j) {
                    float v = Gw[i][j];
                    for (int p = 0; p < i; ++p) v -= R[p][i] * R[p][j];
                    R[i][j] = v * inv;
                }
            }

            // align: a = sum_i mean_rows[i]*R[i][w-1] / ||r[:-1]||, a = 1/(2wa)
            float mr[4];
            for (int i = 0; i < w - 1; ++i) mr[i] = 0.f;
            for (int j = 0; j < w - 1; ++j) {
                float cn2 = 0.f;
                for (int i = 0; i < w - 1; ++i) cn2 += R[i][j] * R[i][j];
                const float icn = 1.f / sqrtf(cn2);
                for (int i = 0; i < w - 1; ++i) mr[i] += R[i][j] * icn;
            }
            float num = 0.f, den2 = 0.f;
            for (int i = 0; i < w - 1; ++i) {
                num  += (mr[i] / (float)(w - 1)) * R[i][w - 1];
                den2 += R[i][w - 1] * R[i][w - 1];
            }
            float a = num / sqrtf(den2);
            a = 1.f / (a * (float)(2 * w));
            align9[k] = a;
            const float rn2 = den2 + R[w - 1][w - 1] * R[w - 1][w - 1];
            nov9[k] = fabsf(R[w - 1][w - 1]) / sqrtf(rn2);
        }

        float sa = 0.f, sn = 0.f;
        for (int k = 0; k < L; ++k) { sa += align9[k]; sn += nov9[k]; }
        float alpha[L], st = 0.f;
        for (int k = 0; k < L; ++k) {
            alpha[k] = align9[k] / sa + nov9[k] / sn;
            st += alpha[k];
        }
        float* aout = &alpha_out[(size_t)tok * L];
        for (int k = 0; k < L; ++k) aout[k] = alpha[k] / st;

        // cosine similarity of adjacent layers -> sample variance (ddof=1)
        float c[L - 1], mean = 0.f;
        for (int j = 0; j < L - 1; ++j) {
            const float nn = sqrtf(s_G[j * 9 + j]) * sqrtf(s_G[(j + 1) * 9 + j + 1]);
            c[j] = s_G[(j + 1) * 9 + j] / fmaxf(nn, 1e-8f);
            mean += c[j];
        }
        mean /= (float)(L - 1);
        float var = 0.f;
        for (int j = 0; j < L - 1; ++j) {
            const float d = c[j] - mean;
            var += d * d;
        }
        var_out[tok] = var / (float)(L - 2);
    }
}

// ---------------------------------------------------------------------------
// Kernel 2: per-batch mask + normalize var_token (in place). Tiny.
// ---------------------------------------------------------------------------
__global__ void wk_norm_kernel(const int* __restrict__ mask,
                               float* __restrict__ var_tok)
{
    const int b = threadIdx.x;
    if (b >= BB) return;
    int um = -1;
    for (int s = 0; s < SS; ++s) um += mask[b * SS + s];
    float* v = &var_tok[b * SS];
    float sum = 0.f;
    for (int s = 0; s < SS; ++s) sum += (s < um) ? v[s] : 0.f;
    const float inv = 1.f / sum;
    for (int s = 0; s < SS; ++s) v[s] = (s < um) ? v[s] * inv : 0.f;
}

// ---------------------------------------------------------------------------
// Kernel 3: out[b][d] = sum_s varn[b][s] * sum_l alpha[l] * feat[l][b][s][d].
// Deterministic (no atomics): block = (b, 128-wide d chunk), loop over s.
// ---------------------------------------------------------------------------
__global__ __launch_bounds__(128) void wk_pool_kernel(
    const float* __restrict__ hs,
    const float* __restrict__ var_tok,
    const float* __restrict__ alpha,
    float* __restrict__ out)
{
    const int b = blockIdx.x;
    const int d = blockIdx.y * 128 + threadIdx.x;
    const size_t layer_stride = (size_t)BB * SS * DD;
    float acc = 0.f;
    for (int s = 0; s < SS; ++s) {
        const float wv = var_tok[b * SS + s];   // uniform per block
        if (wv == 0.f) continue;                 // weight-0 terms contribute 0
        const float* ap = &alpha[(size_t)(b * SS + s) * L];
        const float* fp = &hs[(size_t)LAYER0 * layer_stride +
                              ((size_t)b * SS + s) * DD + d];
        float e = 0.f;
        #pragma unroll
        for (int l = 0; l < L; ++l) e += ap[l] * fp[(size_t)l * layer_stride];
        acc += wv * e;
    }
    out[(size_t)b * DD + d] = acc;
}

// ---------------------------------------------------------------------------
extern "C" void kernel_launch(void* const* d_in, const int* in_sizes, int n_in,
                              void* d_out, int out_size, void* d_ws, size_t ws_size,
                              hipStream_t stream)
{
    const float* hs   = (const float*)d_in[0];   // (13,8,512,1024) fp32
    const int*   mask = (const int*)d_in[1];     // (8,512) int
    float*       out  = (float*)d_out;           // (8,1024) fp32

    float* wsf      = (float*)d_ws;
    float* var_tok  = wsf;                        // B*S floats
    float* alpha_ws = wsf + BB * SS;              // B*S*9 floats (~165 KB total)

    wk_token_kernel<<<BB * SS, 128, 0, stream>>>(hs, var_tok, alpha_ws);
    wk_norm_kernel<<<1, 32, 0, stream>>>(mask, var_tok);
    wk_pool_kernel<<<dim3(BB, DD / 128), 128, 0, stream>>>(hs, var_tok, alpha_ws, out);
}